// TriDiagonalLaplaceSolver_9887014715997
// MI455X (gfx1250) — compile-verified
//
#include <hip/hip_runtime.h>
#include <stdint.h>

// Tridiagonal implicit-Laplace solve (Thomas w/ precomputed LU), B=4096 rows x N=8192.
// Bandwidth-bound (256 MB compulsory @ 23.3 TB/s ~= 11 us). Blocked affine scan
// (segments of 128) -> 8192 waves; row-shared coefficient slices staged into LDS with
// the CDNA5 Tensor Data Mover (tensor_load_to_lds) and async global->LDS copies;
// y0/u0 staged through d_out so the 192 MB L2 absorbs intermediate traffic.

#define NN 8192
#define BB 4096
#define SS 64          // segments per row
#define LL 128         // segment length = NN/SS
#define TPB 256        // threads per block (one row per thread in sweep kernels)

typedef uint32_t u32x4 __attribute__((ext_vector_type(4)));
typedef uint32_t u32x8 __attribute__((ext_vector_type(8)));

// ---------- CDNA5 data-mover helpers (gfx1250) ----------
__device__ __forceinline__ uint32_t lds_off_of(const void* p) {
  // LDS-aperture flat pointers: low 32 bits are the wave-relative LDS byte address.
  return (uint32_t)(uintptr_t)p;
}
__device__ __forceinline__ void async_g2l_b32(uint32_t lds_off, const float* g) {
  asm volatile("global_load_async_to_lds_b32 %0, %1, off"
               :: "v"(lds_off), "v"(g) : "memory");
}
__device__ __forceinline__ void wait_async0() {
#if defined(__has_builtin)
#if __has_builtin(__builtin_amdgcn_s_wait_asynccnt)
  __builtin_amdgcn_s_wait_asynccnt(0);
#else
  asm volatile("s_wait_asynccnt 0x0" ::: "memory");
#endif
#else
  asm volatile("s_wait_asynccnt 0x0" ::: "memory");
#endif
}
__device__ __forceinline__ void wait_tensor0() {
#if defined(__has_builtin)
#if __has_builtin(__builtin_amdgcn_s_wait_tensorcnt)
  __builtin_amdgcn_s_wait_tensorcnt(0);
#else
  asm volatile("s_wait_tensorcnt 0x0" ::: "memory");
#endif
#else
  asm volatile("s_wait_tensorcnt 0x0" ::: "memory");
#endif
}

// TDM: 1D tile of n fp32 from global -> LDS. D# per CDNA5 ISA ch.8 (2-group form,
// VADDR2/3 = NULL). Issue from a single wave (EXEC is ignored by TDM; TENSORcnt is
// per-wave, so only the issuing wave waits non-trivially).
__device__ __forceinline__ void tdm_load_f32_1d(uint32_t lds_off, const float* g,
                                                uint32_t n) {
  uint64_t ga = (uint64_t)(uintptr_t)g;
  u32x4 g0;
  g0[0] = 1u;                                             // count=1, user-mode load
  g0[1] = lds_off;                                        // lds_addr (bytes)
  g0[2] = (uint32_t)ga;                                   // global_addr[31:0]
  g0[3] = ((uint32_t)(ga >> 32) & 0x01FFFFFFu)            // global_addr[56:32]
          | 0x80000000u;                                  // type=2 ("image")
  u32x8 g1;
  g1[0] = 2u << 16;                                       // data_size=2 (4 bytes)
  g1[1] = (n & 0xFFFFu) << 16;                            // tensor_dim0[15:0]
  g1[2] = (n >> 16) | (1u << 16);                         // tensor_dim0[31:16] | tensor_dim1 lo=1
  g1[3] = (n & 0xFFFFu) << 16;                            // tensor_dim1 hi=0 | tile_dim0=n
  g1[4] = 0u;                                             // tile_dim1=0, tile_dim2=0
  g1[5] = n;                                              // tensor_dim0_stride lo
  g1[6] = 0u;
  g1[7] = 0u;
  asm volatile("tensor_load_to_lds %0, %1" :: "s"(g0), "s"(g1) : "memory");
}

// ---------- K0: row-independent coefficient precompute ----------
// P[i]   : forward inflow multiplier within segment,  P = prod(-cp)
// Q[i]   : backward inflow multiplier within segment, Q = prod(-bp/a)
// invA[i]: 1/a[i]; mF[s]=P[last of seg], mB[s]=Q[first of seg]
__global__ void k0_coeff(const float* __restrict__ a, const float* __restrict__ b,
                         const float* __restrict__ c,
                         float* __restrict__ P, float* __restrict__ Q,
                         float* __restrict__ invA, float* __restrict__ mF,
                         float* __restrict__ mB) {
  int t = threadIdx.x;
  if (t < SS) {                       // forward segment t
    int base = t * LL;
    float p = (base == 0) ? 0.f : -c[base - 1];   // cp[0] == 0
    P[base] = p;
    for (int i = 1; i < LL; ++i) {
      p = (-c[base + i - 1]) * p;
      P[base + i] = p;
    }
    mF[t] = p;
  } else if (t < 2 * SS) {            // backward segment s (+ invA for that slice)
    int s = t - SS;
    int base = s * LL;
    int i = base + LL - 1;
    float ia = 1.0f / a[i];
    invA[i] = ia;
    float bp = (i == NN - 1) ? 0.f : b[i];        // bp[N-1] == 0
    float q = -bp * ia;
    Q[i] = q;
    for (i = base + LL - 2; i >= base; --i) {
      ia = 1.0f / a[i];
      invA[i] = ia;
      q = (-b[i] * ia) * q;
      Q[i] = q;
    }
    mB[s] = q;
  }
}

// ---------- K1: forward partial scan (zero inflow), y0 -> d_out ----------
__global__ void k1_fwd_partial(const float* __restrict__ x, const float* __restrict__ c,
                               float* __restrict__ y0, float* __restrict__ tF) {
  __shared__ float4 shc4[LL / 4];
  float* shc = (float*)shc4;
  const int s = blockIdx.x;
  const int base = s * LL;
  const int t = threadIdx.x;
  if (t == 0) {
    // cp[i] = c[i-1]; for s==0 shift by one element and zero cp[0] explicitly
    // (LDS garbage * 0 could be NaN -> must be a finite value).
    const float* src = (s == 0) ? c : (c + base - 1);
    uint32_t dst = lds_off_of(shc) + ((s == 0) ? 4u : 0u);
    uint32_t n = (s == 0) ? (LL - 1) : LL;
    tdm_load_f32_1d(dst, src, n);
    if (s == 0) shc[0] = 0.f;
  }
  wait_tensor0();
  __syncthreads();

  const int row = blockIdx.y * TPB + t;
  const size_t off = ((size_t)row << 13) + base;
  const float4* xp = (const float4*)(x + off);
  float4* yp = (float4*)(y0 + off);
  float yprev = 0.f;                  // partial scan: zero inflow
#pragma unroll
  for (int j = 0; j < LL / 4; ++j) {
    float4 v = xp[j];
    float4 cc = shc4[j];
    float4 r;
    r.x = fmaf(-cc.x, yprev, v.x);
    r.y = fmaf(-cc.y, r.x, v.y);
    r.z = fmaf(-cc.z, r.y, v.z);
    r.w = fmaf(-cc.w, r.z, v.w);
    yprev = r.w;
    yp[j] = r;
  }
  tF[(size_t)s * BB + row] = yprev;   // carry: y0 at segment end (coalesced)
}

// ---------- K2: per-row forward carry resolve (64 sequential steps) ----------
__global__ void k2_fwd_carry(const float* __restrict__ tF, const float* __restrict__ mF,
                             float* __restrict__ yin) {
  __shared__ float shm[SS];
  int t = threadIdx.x;
  if (t < SS) async_g2l_b32(lds_off_of(&shm[t]), mF + t);  // waves 0-1: full EXEC
  wait_async0();
  __syncthreads();
  int row = blockIdx.x * TPB + t;
  float cur = 0.f;
#pragma unroll
  for (int s2 = 0; s2 < SS; ++s2) {
    size_t o = (size_t)s2 * BB + row;             // coalesced
    yin[o] = cur;                                 // inflow of segment s2
    cur = fmaf(shm[s2], cur, tF[o]);              // inflow of segment s2+1
  }
}

// ---------- K3: fused forward fixup + backward partial (streaming), u0 -> d_out ------
__global__ void k3_fwd_fix_bwd_partial(float* __restrict__ y,   // in: y0, out: u0
                                       const float* __restrict__ b,
                                       const float* __restrict__ P,
                                       const float* __restrict__ invA,
                                       const float* __restrict__ yin,
                                       float* __restrict__ tB) {
  __shared__ float4 shP4[LL / 4];
  __shared__ float4 shIA4[LL / 4];
  __shared__ float4 shB4[LL / 4];
  float* shB = (float*)shB4;
  const int s = blockIdx.x;
  const int base = s * LL;
  const int t = threadIdx.x;
  if (t == 0) {                                        // wave 0: P + invA via TDM
    tdm_load_f32_1d(lds_off_of(shP4), P + base, LL);
    tdm_load_f32_1d(lds_off_of(shIA4), invA + base, LL);
  } else if (t >= 64 && t < 64 + LL) {                 // waves 2..5: b slice, clamped
    int j = t - 64;
    int gi = base + j;
    int src = (gi > NN - 2) ? (NN - 2) : gi;           // bp[N-1] only ever multiplied by 0
    async_g2l_b32(lds_off_of(&shB[j]), b + src);
  }
  wait_tensor0();
  wait_async0();
  __syncthreads();

  const int row = blockIdx.y * TPB + t;
  const size_t off = ((size_t)row << 13) + base;
  float4* yp = (float4*)(y + off);
  const float yv = yin[(size_t)s * BB + row];

  float un = 0.f;                                      // backward partial: zero inflow
#pragma unroll
  for (int j = LL / 4 - 1; j >= 0; --j) {
    float4 v = yp[j];                                  // y0
    float4 p = shP4[j];
    float4 bb = shB4[j];
    float4 ia = shIA4[j];
    v.x = fmaf(p.x, yv, v.x);                          // y = y0 + P*y_in (elementwise)
    v.y = fmaf(p.y, yv, v.y);
    v.z = fmaf(p.z, yv, v.z);
    v.w = fmaf(p.w, yv, v.w);
    v.w = fmaf(-bb.w, un, v.w) * ia.w;                 // u = (y - b*u_next) * invA
    v.z = fmaf(-bb.z, v.w, v.z) * ia.z;
    v.y = fmaf(-bb.y, v.z, v.y) * ia.y;
    v.x = fmaf(-bb.x, v.y, v.x) * ia.x;
    un = v.x;
    yp[j] = v;                                         // overwrite with u0
  }
  tB[(size_t)s * BB + row] = un;                       // carry: u0 at segment start
}

// ---------- K4: per-row backward carry resolve ----------
__global__ void k4_bwd_carry(const float* __restrict__ tB, const float* __restrict__ mB,
                             float* __restrict__ zin) {
  __shared__ float shm[SS];
  int t = threadIdx.x;
  if (t < SS) async_g2l_b32(lds_off_of(&shm[t]), mB + t);
  wait_async0();
  __syncthreads();
  int row = blockIdx.x * TPB + t;
  float cur = 0.f;                                     // inflow of last segment
#pragma unroll
  for (int s2 = SS - 1; s2 >= 0; --s2) {
    size_t o = (size_t)s2 * BB + row;                  // coalesced
    zin[o] = cur;                                      // inflow of segment s2
    cur = fmaf(shm[s2], cur, tB[o]);                   // inflow of segment s2-1
  }
}

// ---------- K5: backward fixup, z = u0 + Q * z_in (streaming) ----------
__global__ void k5_bwd_fix(float* __restrict__ z, const float* __restrict__ Q,
                           const float* __restrict__ zin) {
  __shared__ float4 shQ4[LL / 4];
  const int s = blockIdx.x;
  const int base = s * LL;
  const int t = threadIdx.x;
  if (t == 0) tdm_load_f32_1d(lds_off_of(shQ4), Q + base, LL);
  wait_tensor0();
  __syncthreads();
  const int row = blockIdx.y * TPB + t;
  const size_t off = ((size_t)row << 13) + base;
  float4* zp = (float4*)(z + off);
  const float zv = zin[(size_t)s * BB + row];
#pragma unroll
  for (int j = 0; j < LL / 4; ++j) {
    float4 v = zp[j];
    float4 q = shQ4[j];
    v.x = fmaf(q.x, zv, v.x);
    v.y = fmaf(q.y, zv, v.y);
    v.z = fmaf(q.z, zv, v.z);
    v.w = fmaf(q.w, zv, v.w);
    zp[j] = v;
  }
}

extern "C" void kernel_launch(void* const* d_in, const int* in_sizes, int n_in,
                              void* d_out, int out_size, void* d_ws, size_t ws_size,
                              hipStream_t stream) {
  (void)in_sizes; (void)n_in; (void)out_size; (void)ws_size;
  const float* x = (const float*)d_in[0];
  const float* a = (const float*)d_in[1];
  const float* b = (const float*)d_in[2];
  const float* c = (const float*)d_in[3];
  float* out = (float*)d_out;

  // workspace layout (floats): ~4.1 MB total
  float* w    = (float*)d_ws;
  float* P    = w;                          // NN
  float* Q    = P + NN;                     // NN
  float* invA = Q + NN;                     // NN
  float* mF   = invA + NN;                  // SS
  float* mB   = mF + SS;                    // SS
  float* tF   = mB + SS;                    // SS*BB  (layout [s][row], coalesced)
  float* yin  = tF + (size_t)SS * BB;       // SS*BB
  float* tB   = yin + (size_t)SS * BB;      // SS*BB
  float* zin  = tB + (size_t)SS * BB;       // SS*BB

  dim3 gseg(SS, BB / TPB);                  // 64 x 16 blocks, 256 thr = 8192 waves
  k0_coeff<<<1, 2 * SS, 0, stream>>>(a, b, c, P, Q, invA, mF, mB);
  k1_fwd_partial<<<gseg, TPB, 0, stream>>>(x, c, out, tF);
  k2_fwd_carry<<<BB / TPB, TPB, 0, stream>>>(tF, mF, yin);
  k3_fwd_fix_bwd_partial<<<gseg, TPB, 0, stream>>>(out, b, P, invA, yin, tB);
  k4_bwd_carry<<<BB / TPB, TPB, 0, stream>>>(tB, mB, zin);
  k5_bwd_fix<<<gseg, TPB, 0, stream>>>(out, Q, zin);
}